// ProteinGCN_41532333752332
// MI455X (gfx1250) — compile-verified
//
#include <hip/hip_runtime.h>
#include <math.h>

typedef __attribute__((ext_vector_type(2))) float v2f;
typedef __attribute__((ext_vector_type(8))) float v8f;

#define F1 64
#define F2 128

// ---------------------------------------------------------------------------
// K1: zero the degree array
__global__ void k_zero_deg(float* __restrict__ deg, int N) {
    int i = blockIdx.x * blockDim.x + threadIdx.x;
    if (i < N) deg[i] = 0.0f;
}

// K2: count in-degree per target node (self-loop added later as +1)
__global__ void k_count_deg(const int* __restrict__ ei, float* __restrict__ deg,
                            int E) {
    int e = blockIdx.x * blockDim.x + threadIdx.x;
    if (e < E) {
        int c = ei[E + e];  // target
        unsafeAtomicAdd(&deg[c], 1.0f);
    }
}

// K3: dinv = rsqrt(deg + 1); init scalar aggregate with self-loop term
__global__ void k_dinv_init(const float* __restrict__ deg,
                            const float* __restrict__ x,
                            float* __restrict__ dinv, float* __restrict__ sagg,
                            int N) {
    int i = blockIdx.x * blockDim.x + threadIdx.x;
    if (i < N) {
        float d  = deg[i] + 1.0f;        // self loop
        float di = rsqrtf(d);            // deg>0 always (self loop)
        dinv[i]  = di;
        sagg[i]  = di * di * x[i];       // self-loop message (feature dim = 1)
    }
}

// K4: layer-1 edge scatter (scalar because F_in == 1)
__global__ void k_edge_l1(const int* __restrict__ ei,
                          const float* __restrict__ x,
                          const float* __restrict__ dinv,
                          float* __restrict__ sagg, int E) {
    int e = blockIdx.x * blockDim.x + threadIdx.x;
    if (e < E) {
        int r = ei[e];
        int c = ei[E + e];
        float nrm = dinv[r] * dinv[c];
        unsafeAtomicAdd(&sagg[c], nrm * x[r]);
    }
}

// K5: h2[N,128] = relu(sagg[:,None]*W1 + b1) @ W2   via V_WMMA_F32_16X16X4_F32
// Block = 256 threads = 8 waves; each wave owns one 16x16 output tile,
// block covers 16 rows x 128 cols. 16 K-steps of 4.
__global__ void __launch_bounds__(256)
k_gemm_wmma(const float* __restrict__ sagg, const float* __restrict__ W1,
            const float* __restrict__ b1, const float* __restrict__ W2,
            float* __restrict__ h2, int N) {
    const int wave = threadIdx.x >> 5;           // 0..7 -> column tile
    const int lane = threadIdx.x & 31;
    const int half = lane >> 4;                  // 0: K=0,1  1: K=2,3
    const int l    = lane & 15;
    const int m0   = blockIdx.x * 16;
    const int n0   = wave * 16;

    int rowA = m0 + l;
    if (rowA > N - 1) rowA = N - 1;              // clamp (keeps EXEC uniform)
    const float s = sagg[rowA];

    v8f acc = {};
    #pragma unroll
    for (int kk = 0; kk < F1; kk += 4) {
        const int k0 = kk + half * 2;
        const int k1 = k0 + 1;
        v2f a, b;
        a.x = fmaxf(fmaf(s, W1[k0], b1[k0]), 0.0f);   // relu(s*W1+b1)
        a.y = fmaxf(fmaf(s, W1[k1], b1[k1]), 0.0f);
        b.x = W2[k0 * F2 + n0 + l];
        b.y = W2[k1 * F2 + n0 + l];
        acc = __builtin_amdgcn_wmma_f32_16x16x4_f32(
            /*neg_a=*/false, a, /*neg_b=*/false, b,
            /*c_mod=*/(short)0, acc, /*reuse_a=*/false, /*reuse_b=*/false);
    }

    // D layout: VGPR p -> row m0 + p + half*8, col n0 + l
    #pragma unroll
    for (int p = 0; p < 8; ++p) {
        int row = m0 + p + half * 8;
        if (row < N) h2[row * F2 + n0 + l] = acc[p];
    }
}

// K6: out = dinv^2 * h2 (self loop) + b2   (full overwrite -> re-inits d_out)
__global__ void k_init_out(const float* __restrict__ h2,
                           const float* __restrict__ dinv,
                           const float* __restrict__ b2,
                           float* __restrict__ out, int N) {
    int idx = blockIdx.x * blockDim.x + threadIdx.x;
    if (idx < N * F2) {
        int c = idx >> 7;           // /128
        int k = idx & (F2 - 1);
        float di = dinv[c];
        out[idx] = di * di * h2[idx] + b2[k];
    }
}

// K7: layer-2 edge scatter. One wave (32 lanes) per edge; 4 features/lane.
__global__ void k_edge_l2(const int* __restrict__ ei,
                          const float* __restrict__ dinv,
                          const float* __restrict__ h2,
                          float* __restrict__ out, int E) {
    int t = blockIdx.x * blockDim.x + threadIdx.x;
    int e = t >> 5;
    int lane = t & 31;
    if (e < E) {
        int r = ei[e];
        int c = ei[E + e];
        float nrm = dinv[r] * dinv[c];
        int k = lane * 4;
        const float4 hv = *(const float4*)(h2 + r * F2 + k);
        float* dst = out + c * F2 + k;
        unsafeAtomicAdd(dst + 0, nrm * hv.x);
        unsafeAtomicAdd(dst + 1, nrm * hv.y);
        unsafeAtomicAdd(dst + 2, nrm * hv.z);
        unsafeAtomicAdd(dst + 3, nrm * hv.w);
    }
}

// K8: in-place log_softmax over 128 features; one wave32 per row.
__global__ void __launch_bounds__(256)
k_log_softmax(float* __restrict__ out, int N) {
    const int wave = threadIdx.x >> 5;
    const int lane = threadIdx.x & 31;
    const int row  = blockIdx.x * 8 + wave;
    if (row >= N) return;
    float* p = out + row * F2;

    float v[4];
    #pragma unroll
    for (int j = 0; j < 4; ++j) v[j] = p[lane + 32 * j];

    float m = fmaxf(fmaxf(v[0], v[1]), fmaxf(v[2], v[3]));
    #pragma unroll
    for (int off = 16; off > 0; off >>= 1)
        m = fmaxf(m, __shfl_xor(m, off, 32));

    float sum = 0.0f;
    #pragma unroll
    for (int j = 0; j < 4; ++j) sum += __expf(v[j] - m);
    #pragma unroll
    for (int off = 16; off > 0; off >>= 1)
        sum += __shfl_xor(sum, off, 32);

    const float lse = m + __logf(sum);
    #pragma unroll
    for (int j = 0; j < 4; ++j) p[lane + 32 * j] = v[j] - lse;
}

// ---------------------------------------------------------------------------
extern "C" void kernel_launch(void* const* d_in, const int* in_sizes, int n_in,
                              void* d_out, int out_size, void* d_ws, size_t ws_size,
                              hipStream_t stream) {
    const float* x  = (const float*)d_in[0];
    const int*   ei = (const int*)d_in[1];     // [2,E]: row 0 = src, row 1 = dst
    const float* W1 = (const float*)d_in[2];   // [1,64]
    const float* b1 = (const float*)d_in[3];   // [64]
    const float* W2 = (const float*)d_in[4];   // [64,128]
    const float* b2 = (const float*)d_in[5];   // [128]
    float*       out = (float*)d_out;          // [N,128]

    const int N = in_sizes[0];                 // x is [N,1]
    const int E = in_sizes[1] / 2;

    // workspace layout
    float* deg  = (float*)d_ws;                // N
    float* dinv = deg + N;                     // N
    float* sagg = dinv + N;                    // N
    float* h2   = sagg + N;                    // N*128

    const int B = 256;
    auto cdiv = [](long long a, long long b) { return (int)((a + b - 1) / b); };

    k_zero_deg  <<<cdiv(N, B), B, 0, stream>>>(deg, N);
    k_count_deg <<<cdiv(E, B), B, 0, stream>>>(ei, deg, E);
    k_dinv_init <<<cdiv(N, B), B, 0, stream>>>(deg, x, dinv, sagg, N);
    k_edge_l1   <<<cdiv(E, B), B, 0, stream>>>(ei, x, dinv, sagg, E);
    k_gemm_wmma <<<cdiv(N, 16), B, 0, stream>>>(sagg, W1, b1, W2, h2, N);
    k_init_out  <<<cdiv((long long)N * F2, B), B, 0, stream>>>(h2, dinv, b2, out, N);
    k_edge_l2   <<<cdiv((long long)E * 32, B), B, 0, stream>>>(ei, dinv, h2, out, E);
    k_log_softmax<<<cdiv(N, 8), B, 0, stream>>>(out, N);
}